// NeuSRenderer_48447231099462
// MI455X (gfx1250) — compile-verified
//
#include <hip/hip_runtime.h>
#include <stdint.h>

// ---------------- problem constants (from reference) ----------------
#define NVIEW     32
#define NRC       512          // coarse rays / view (64 x 8)
#define NRF       576          // fine rays / view (512 + 64 pdf)
#define NPTS_C    257          // coarse sdf sample points per ray
#define NMID_C    256
#define NEDGE_F   289          // 257 base + 32 pdf steps
#define NMID_F    288
#define NBINS     128
#define BINSZ     0.015f
#define HIDDEN    64
#define MINCOS    0.8660254037844387f       // cos(FOV*pi/360), FOV=60
#define SOLIDANG  (6.283185307179586f * (1.0f - MINCOS))
#define INVS_C    64.0f

typedef float v2f __attribute__((ext_vector_type(2)));
typedef float v8f __attribute__((ext_vector_type(8)));

// ---------------- small device helpers ----------------
static __device__ __forceinline__ uint32_t pcg(uint32_t v) {
  v = v * 747796405u + 2891336453u;
  v = ((v >> ((v >> 28u) + 4u)) ^ v) * 277803737u;
  return (v >> 22u) ^ v;
}
static __device__ __forceinline__ float u01(uint32_t h) {
  return (float)(h >> 8) * (1.0f / 16777216.0f);
}
static __device__ __forceinline__ uint32_t keymix(uint32_t s, uint32_t a, uint32_t b, uint32_t c) {
  return pcg(pcg(pcg(s ^ 0x9E3779B9u) + a * 0x85EBCA6Bu) ^ (b * 0xC2B2AE35u)) + pcg(c + 0x27D4EB2Fu);
}
// fast activations: branch-free, hit the v_exp_f32 / v_log_f32 transcendental
// pipe (co-executes with the WMMA/XDL pipe) instead of the libm expansions.
static __device__ __forceinline__ float sigm(float x) {
  return __fdividef(1.0f, 1.0f + __expf(-x));
}
static __device__ __forceinline__ float softplus(float x) {
  // max(x,0) + log(1 + exp(-|x|)) : stable for all x, no branches
  return fmaxf(x, 0.0f) + __logf(1.0f + __expf(-fabsf(x)));
}
static __device__ __forceinline__ float wave_sum(float x) {
  for (int m = 16; m >= 1; m >>= 1) x += __shfl_xor(x, m, 32);
  return x;
}

// D = A(16x4) * B(4x16) + 0, fp32 exact  -> v_wmma_f32_16x16x4_f32
static __device__ __forceinline__ v8f wmma4(v2f a, v2f b) {
  v8f c = {};
  return __builtin_amdgcn_wmma_f32_16x16x4_f32(false, a, false, b, (short)0, c, false, false);
}

// load the constant A-operand (hidden-on-M weight tiles) once per wave.
// A[m][k] tile t: rows m=16t..16t+15 hold (W1[0][h], W1[1][h] | W1[2][h], b1[h])
static __device__ __forceinline__ void load_wmma_A(const float* __restrict__ W1,
                                                   const float* __restrict__ b1,
                                                   int lane, v2f aW[4]) {
  int li = lane & 15;
  bool lo = lane < 16;
#pragma unroll
  for (int t = 0; t < 4; ++t) {
    int h = 16 * t + li;
    aW[t].x = lo ? W1[0 * HIDDEN + h] : W1[2 * HIDDEN + h];
    aW[t].y = lo ? W1[1 * HIDDEN + h] : b1[h];
  }
}

// sample one camera ray (matches sample_rays structure; RNG is our own hash)
static __device__ __forceinline__ void make_ray(const float* __restrict__ Rt, int v, int slot,
                                                int uvi, uint32_t salt,
                                                float& ox, float& oy, float& oz,
                                                float& dx, float& dy, float& dz, float& cos_t) {
  int uu = uvi >> 3, vv = uvi & 7;
  float ru = ((float)uu + u01(keymix(salt, (uint32_t)v, (uint32_t)slot, 1u))) * (1.0f / 64.0f);
  float rv = ((float)vv + u01(keymix(salt, (uint32_t)v, (uint32_t)slot, 2u))) * (1.0f / 8.0f);
  float phi = ru * 6.283185307179586f;
  float ct = rv * (1.0f - MINCOS) + MINCOS;
  float st = sqrtf(fmaxf(0.0f, 1.0f - ct * ct));
  float l0 = st * __cosf(phi), l1 = st * __sinf(phi), l2 = ct;
  const float* R = Rt + v * 12;  // (3,4) row-major
  ox = R[3]; oy = R[7]; oz = R[11];
  dx = l0 * R[0] + l1 * R[1] + l2 * R[2];
  dy = l0 * R[4] + l1 * R[5] + l2 * R[6];
  dz = l0 * R[8] + l1 * R[9] + l2 * R[10];
  cos_t = ct;
}

// stratified coarse/fine base step position for sample index s (0..256)
static __device__ __forceinline__ float base_step(int s, int v, int r, uint32_t salt) {
  if (s >= 256) return 128.0f * BINSZ;
  int bin = s >> 1;
  float t = (float)bin * BINSZ;
  if (s & 1) t += u01(keymix(salt, (uint32_t)v, (uint32_t)r, (uint32_t)bin)) * BINSZ;
  return t;
}

// ============================================================================
// Kernel 1: coarse pass.  1 wave = 1 ray. SDF MLP via v_wmma_f32_16x16x4_f32.
// ============================================================================
__global__ void __launch_bounds__(128) neus_coarse_kernel(
    const float* __restrict__ Rt, const float* __restrict__ W1, const float* __restrict__ b1,
    const float* __restrict__ w2, const float* __restrict__ b2,
    float* __restrict__ ray_pdf, float* __restrict__ step_pdf) {
  __shared__ float wtbl[HIDDEN];      // w2
  __shared__ float tS[4][NPTS_C + 3];
  __shared__ float sS[4][NPTS_C + 3];

  __builtin_prefetch(W1, 0, 3);
  __builtin_prefetch(w2, 0, 3);

  int tid = threadIdx.x, lane = tid & 31, wid = tid >> 5;
  if (tid < HIDDEN) wtbl[tid] = w2[tid];
  __syncthreads();

  int gid = blockIdx.x * 4 + wid;
  int v = gid >> 9, r = gid & (NRC - 1);
  float b2v = b2[0];

  float ox, oy, oz, dx, dy, dz, ct;
  make_ray(Rt, v, r, r, 0x10u, ox, oy, oz, dx, dy, dz, ct);

  v2f aW[4];
  load_wmma_A(W1, b1, lane, aW);
  int hi = lane >> 4;

  // ---- pass A: SDF at 257 step points, 16 points per WMMA group ----
  for (int c = 0; c < 17; ++c) {
    int p = lane & 15;
    int s = 16 * c + p;
    int sc = s > 256 ? 256 : s;
    float t = base_step(sc, v, r, 0x11u);
    float px = ox + t * dx, py = oy + t * dy, pz = oz + t * dz;
    v2f bOp;
    bOp.x = (lane < 16) ? px : pz;
    bOp.y = (lane < 16) ? py : 1.0f;
    float acc_s = 0.0f;
#pragma unroll
    for (int tt = 0; tt < 4; ++tt) {
      v8f z = wmma4(aW[tt], bOp);
#pragma unroll
      for (int q = 0; q < 8; ++q) {
        int h = 16 * tt + 8 * hi + q;
        acc_s += softplus(z[q]) * wtbl[h];
      }
    }
    float sdf = acc_s + __shfl_xor(acc_s, 16, 32) + b2v;
    if (lane < 16 && s <= 256) { tS[wid][s] = t; sS[wid][s] = sdf; }
  }

  // ---- pass B: alpha + transmittance scan + pdf accumulation ----
  float carryT = 1.0f, aPrev = 0.0f, raySum = 0.0f;
  for (int pass = 0; pass < 8; ++pass) {
    int i = 32 * pass + lane;
    float s0 = sS[wid][i], s1 = sS[wid][i + 1];
    float t0 = tS[wid][i], t1 = tS[wid][i + 1];
    float delta = t1 - t0;
    float msdf = 0.5f * (s0 + s1);
    float mt = 0.5f * (t0 + t1);
    float mx = ox + mt * dx, my = oy + mt * dy, mz = oz + mt * dz;
    float inside = (sqrtf(mx * mx + my * my + mz * mz) < 1.0f && mz > 0.0f) ? 1.0f : 0.0f;
    float cosv = __fdividef(s1 - s0, delta + 1e-6f);
    float pcos = 0.0f;
    if (i > 0) {
      float sm = sS[wid][i - 1], tm = tS[wid][i - 1];
      pcos = __fdividef(s0 - sm, t0 - tm + 1e-6f);
    }
    float cc = fminf(pcos, cosv);
    cc = fminf(fmaxf(cc, -1000.0f), 0.0f);
    float dsdf = 0.5f * cc * delta;
    float pT = sigm(INVS_C * (msdf - dsdf));
    float nT = sigm(INVS_C * (msdf + dsdf));
    float a = (1.0f - __fdividef(nT, pT + 1e-6f)) * inside;

    float ap = __shfl_up(a, 1, 32);
    float f;
    if (lane == 0) f = (i == 0) ? 1.0f : ((1.0f - aPrev) * (1.0f - aPrev) + 1e-6f);
    else           f = (1.0f - ap) * (1.0f - ap) + 1e-6f;
    float incl = f;
#pragma unroll
    for (int off = 1; off < 32; off <<= 1) {
      float o2 = __shfl_up(incl, off, 32);
      if (lane >= off) incl *= o2;
    }
    float T = carryT * incl;
    float w = a * T;
    raySum += w;
    float wpair = w + __shfl_xor(w, 1, 32);
    if (!(lane & 1)) step_pdf[(size_t)(v * NRC + r) * NBINS + (i >> 1)] = wpair;
    carryT *= __shfl(incl, 31, 32);
    aPrev = __shfl(a, 31, 32);
  }
  float tot = wave_sum(raySum);
  if (lane == 0) ray_pdf[v * NRC + r] = tot;
}

// ============================================================================
// Kernel 2: categorical ray sampling (inverse CDF on ray_pdf + 1e-6)
// ============================================================================
__global__ void __launch_bounds__(64) neus_ray_sample_kernel(
    const float* __restrict__ ray_pdf, int* __restrict__ extras, float* __restrict__ cnts) {
  int v = blockIdx.x, j = threadIdx.x;
  const float* p = ray_pdf + v * NRC;
  float tot = 0.0f;
  for (int i = 0; i < NRC; ++i) tot += p[i] + 1e-6f;
  float u = u01(keymix(0x54u, (uint32_t)v, (uint32_t)j, 7u)) * tot;
  float acc = 0.0f;
  int idx = NRC - 1;
  for (int i = 0; i < NRC; ++i) {
    acc += p[i] + 1e-6f;
    if (u < acc) { idx = i; break; }
  }
  extras[v * 64 + j] = idx;
  atomicAdd(&cnts[v * NRC + idx], 1.0f);
}

// ============================================================================
// Kernel 3: fine pass. 1 wave = 1 ray. SDF + gradient via WMMA, sorted-merge
// of pdf steps, transmittance scan, LDS bin scatter, histogram atomics.
// ============================================================================
__global__ void __launch_bounds__(128) neus_fine_kernel(
    const float* __restrict__ Rt, const float* __restrict__ W1, const float* __restrict__ b1,
    const float* __restrict__ w2, const float* __restrict__ b2,
    const float* __restrict__ inv_s_p, const float* __restrict__ rho_p,
    const float* __restrict__ step_pdf, const int* __restrict__ extras,
    const float* __restrict__ cnts,
    float* __restrict__ out_sdf, float* __restrict__ out_nrm, float* __restrict__ out_alpha,
    float* __restrict__ out_T, float* __restrict__ out_w, float* __restrict__ out_ba,
    float* __restrict__ out_bw, float* __restrict__ out_op, float* __restrict__ hists) {
  __shared__ float4 tbl[HIDDEN];      // {w2, w2*W1[0], w2*W1[1], w2*W1[2]}
  __shared__ float edges[4][NEDGE_F + 3];
  __shared__ float exS[4][32];
  __shared__ float alf[4][NMID_F];   // also reused as base-step temp (257 <= 288)
  __shared__ float preS[4][NMID_F];
  __shared__ float binA[4][NBINS];
  __shared__ float binW[4][NBINS];
  __shared__ float binR[4][NBINS];
  __shared__ float spdf[4][NBINS];

  __builtin_prefetch(W1, 0, 3);
  __builtin_prefetch(w2, 0, 3);

  int tid = threadIdx.x, lane = tid & 31, wid = tid >> 5;
  if (tid < HIDDEN) {
    float wv = w2[tid];
    tbl[tid] = make_float4(wv, wv * W1[tid], wv * W1[HIDDEN + tid], wv * W1[2 * HIDDEN + tid]);
  }
  __syncthreads();

  int gid = blockIdx.x * 4 + wid;
  int v = gid / NRF, rr = gid % NRF;
  int uv = (rr < NRC) ? rr : extras[v * 64 + (rr - NRC)];
  float b2v = b2[0], inv_s = inv_s_p[0], rho = rho_p[0];

  float ox, oy, oz, dx, dy, dz, ct;
  make_ray(Rt, v, rr, uv, 0x50u, ox, oy, oz, dx, dy, dz, ct);
  float wray = __fdividef(1.0f, (cnts[v * NRC + uv] + 1.0f) * (float)NRC) * ct * SOLIDANG;

  v2f aW[4];
  load_wmma_A(W1, b1, lane, aW);
  int hi = lane >> 4;

  // ---- per-ray step pdf into LDS ----
  for (int i = lane; i < NBINS; i += 32)
    spdf[wid][i] = step_pdf[(size_t)(v * NRC + uv) * NBINS + i] + 1e-6f;
  float part = 0.0f;
  for (int i = lane; i < NBINS; i += 32) part += spdf[wid][i];
  float ptot = wave_sum(part);

  // ---- 32 pdf-sampled extra steps, one per lane ----
  float u = u01(keymix(0x52u, (uint32_t)v, (uint32_t)rr, (uint32_t)lane)) * ptot;
  float acc = 0.0f;
  int bsel = NBINS - 1;
  for (int i = 0; i < NBINS; ++i) {
    acc += spdf[wid][i];
    if (u < acc) { bsel = i; break; }
  }
  float ex = (float)bsel * BINSZ +
             u01(keymix(0x53u, (uint32_t)v, (uint32_t)rr, (uint32_t)lane)) * BINSZ;
  // bitonic sort the 32 extras across the wave
#pragma unroll
  for (int k = 2; k <= 32; k <<= 1)
#pragma unroll
    for (int j = k >> 1; j > 0; j >>= 1) {
      float pv = __shfl_xor(ex, j, 32);
      bool upb = ((lane & k) == 0);
      bool low = ((lane & j) == 0);
      ex = (upb == low) ? fminf(ex, pv) : fmaxf(ex, pv);
    }
  exS[wid][lane] = ex;

  // ---- base steps + merge-path insert of extras -> sorted edges[289] ----
  for (int s = lane; s < NPTS_C; s += 32) alf[wid][s] = base_step(s, v, rr, 0x51u);
  for (int s = lane; s < NPTS_C; s += 32) {
    float bvv = alf[wid][s];
    int cnt = 0;
    for (int j = 0; j < 32; ++j) cnt += (exS[wid][j] < bvv) ? 1 : 0;
    edges[wid][s + cnt] = bvv;
  }
  {
    int cnt = 0;
    for (int s = 0; s < NPTS_C; ++s) cnt += (alf[wid][s] <= ex) ? 1 : 0;
    edges[wid][lane + cnt] = ex;
  }
  for (int i = lane; i < NBINS; i += 32) { binA[wid][i] = 0.f; binW[wid][i] = 0.f; binR[wid][i] = 0.f; }

  // ---- chunk pass: SDF + gradient at 288 mid points (18 x WMMA group) ----
  for (int c = 0; c < 18; ++c) {
    int p = lane & 15;
    int s = 16 * c + p;
    float e0 = edges[wid][s], e1 = edges[wid][s + 1];
    float tm = 0.5f * (e0 + e1);
    float delta = e1 - e0;
    float px = ox + tm * dx, py = oy + tm * dy, pz = oz + tm * dz;
    float inside = (sqrtf(px * px + py * py + pz * pz) < 1.0f && pz > 0.0f) ? 1.0f : 0.0f;
    v2f bOp;
    bOp.x = (lane < 16) ? px : pz;
    bOp.y = (lane < 16) ? py : 1.0f;
    float acc_s = 0.f, g0 = 0.f, g1 = 0.f, g2 = 0.f;
#pragma unroll
    for (int tt = 0; tt < 4; ++tt) {
      v8f z = wmma4(aW[tt], bOp);
#pragma unroll
      for (int q = 0; q < 8; ++q) {
        int h = 16 * tt + 8 * hi + q;
        float zz = z[q];
        float4 cw = tbl[h];
        acc_s += softplus(zz) * cw.x;
        float sg = sigm(zz);
        g0 += sg * cw.y; g1 += sg * cw.z; g2 += sg * cw.w;
      }
    }
    float sdf = (acc_s + __shfl_xor(acc_s, 16, 32) + b2v) * inside;
    float nx = (g0 + __shfl_xor(g0, 16, 32)) * inside;
    float ny = (g1 + __shfl_xor(g1, 16, 32)) * inside;
    float nz = (g2 + __shfl_xor(g2, 16, 32)) * inside;
    float cosv = dx * nx + dy * ny + dz * nz;
    float ac = -fmaxf(0.0f, 0.5f - 0.5f * cosv);    // COS_ANNEAL = 0
    float dsdf = 0.5f * ac * delta;
    float pT = sigm(inv_s * (sdf - dsdf));
    float nT = sigm(inv_s * (sdf + dsdf));
    float a = (1.0f - __fdividef(nT, pT + 1e-6f)) * inside;
    if (lane < 16) {
      size_t base = (size_t)(v * NRF + rr) * NMID_F + s;
      out_sdf[base] = sdf;
      out_nrm[base * 3 + 0] = nx;
      out_nrm[base * 3 + 1] = ny;
      out_nrm[base * 3 + 2] = nz;
      alf[wid][s] = a;
      preS[wid][s] = rho * (-__fdividef(ac, tm * tm + 1e-6f));
    }
  }

  // ---- scan pass: transmittance, weights, radiance, bin scatter ----
  float carryT = 1.0f, aPrev = 0.0f;
  for (int pass = 0; pass < 9; ++pass) {
    int s = 32 * pass + lane;
    float a = alf[wid][s];
    float ap = __shfl_up(a, 1, 32);
    float f;
    if (lane == 0) f = (s == 0) ? 1.0f : (1.0f - aPrev) * (1.0f - aPrev);
    else           f = (1.0f - ap) * (1.0f - ap);
    float incl = f;
#pragma unroll
    for (int off = 1; off < 32; off <<= 1) {
      float o2 = __shfl_up(incl, off, 32);
      if (lane >= off) incl *= o2;
    }
    float T = carryT * incl;
    float w = a * T;
    float rad = w * preS[wid][s];
    size_t base = (size_t)(v * NRF + rr) * NMID_F + s;
    out_alpha[base] = a;
    out_T[base] = T;
    out_w[base] = w;
    float tm = 0.5f * (edges[wid][s] + edges[wid][s + 1]);
    int bidx = (int)floorf(tm * (1.0f / BINSZ));
    bidx = bidx < 0 ? 0 : (bidx > NBINS - 1 ? NBINS - 1 : bidx);
    bool valid = (tm > (float)bidx * BINSZ) && (tm < (float)(bidx + 1) * BINSZ);
    if (valid) {
      atomicAdd(&binA[wid][bidx], a);
      atomicAdd(&binW[wid][bidx], w);
      atomicAdd(&binR[wid][bidx], rad);
    }
    carryT *= __shfl(incl, 31, 32);
    aPrev = __shfl(a, 31, 32);
  }

  // ---- per-ray outputs ----
  float opart = 0.0f;
  for (int i = lane; i < NBINS; i += 32) {
    float ba = binA[wid][i], bw = binW[wid][i], br = binR[wid][i];
    size_t base = (size_t)(v * NRF + rr) * NBINS + i;
    out_ba[base] = ba;
    out_bw[base] = bw;
    opart += bw;
    atomicAdd(&hists[v * NBINS + i], br * wray);
  }
  float op = wave_sum(opart);
  if (lane == 0) out_op[v * NRF + rr] = op;
}

// ============================================================================
// host launcher
// ============================================================================
extern "C" void kernel_launch(void* const* d_in, const int* in_sizes, int n_in,
                              void* d_out, int out_size, void* d_ws, size_t ws_size,
                              hipStream_t stream) {
  (void)in_sizes; (void)n_in; (void)out_size; (void)ws_size;
  const float* Rt = (const float*)d_in[0];
  const float* W1 = (const float*)d_in[1];
  const float* b1 = (const float*)d_in[2];
  const float* w2 = (const float*)d_in[3];
  const float* b2 = (const float*)d_in[4];
  const float* inv_s = (const float*)d_in[5];
  const float* rho = (const float*)d_in[6];

  float* out = (float*)d_out;
  size_t off = 0;
  float* hists = out + off;  off += (size_t)NVIEW * NBINS;                 // (32,128)
  float* o_sdf = out + off;  off += (size_t)NVIEW * NRF * NMID_F;          // (32,576,288)
  float* o_nrm = out + off;  off += (size_t)NVIEW * NRF * NMID_F * 3;      // (...,3)
  float* o_alp = out + off;  off += (size_t)NVIEW * NRF * NMID_F;
  float* o_T   = out + off;  off += (size_t)NVIEW * NRF * NMID_F;
  float* o_w   = out + off;  off += (size_t)NVIEW * NRF * NMID_F;
  float* o_ba  = out + off;  off += (size_t)NVIEW * NRF * NBINS;           // (32,576,128)
  float* o_bw  = out + off;  off += (size_t)NVIEW * NRF * NBINS;
  float* o_op  = out + off;  off += (size_t)NVIEW * NRF;                   // (32,576)
  float* rpdf  = out + off;  off += (size_t)NVIEW * NRC;                   // (32,64,8)
  float* spdf  = out + off;  off += (size_t)NVIEW * NRC * NBINS;           // (32,512,128)

  int*   extras = (int*)d_ws;                                  // 32*64 ints
  float* cnts   = (float*)((char*)d_ws + 4096);                // 32*512 floats

  hipMemsetAsync(hists, 0, (size_t)NVIEW * NBINS * sizeof(float), stream);
  hipMemsetAsync(cnts, 0, (size_t)NVIEW * NRC * sizeof(float), stream);

  dim3 blk(128);
  neus_coarse_kernel<<<dim3(NVIEW * NRC / 4), blk, 0, stream>>>(Rt, W1, b1, w2, b2, rpdf, spdf);
  neus_ray_sample_kernel<<<dim3(NVIEW), dim3(64), 0, stream>>>(rpdf, extras, cnts);
  neus_fine_kernel<<<dim3(NVIEW * NRF / 4), blk, 0, stream>>>(
      Rt, W1, b1, w2, b2, inv_s, rho, spdf, extras, cnts,
      o_sdf, o_nrm, o_alp, o_T, o_w, o_ba, o_bw, o_op, hists);
}